// ExpanderLinearLayer_85744727097647
// MI455X (gfx1250) — compile-verified
//
#include <hip/hip_runtime.h>

#define IN_F   4096
#define OUT_F  4096
#define NROWS  16384

typedef __attribute__((ext_vector_type(2))) float v2f;
typedef __attribute__((ext_vector_type(8))) float v8f;

// ---------------------------------------------------------------------------
// Kernel 1: zero the dense W^T buffer (64 MB in workspace).
// ---------------------------------------------------------------------------
__global__ void expander_zero_kernel(float4* __restrict__ w, int n4) {
    int i = blockIdx.x * blockDim.x + threadIdx.x;
    int stride = gridDim.x * blockDim.x;
    for (; i < n4; i += stride) w[i] = make_float4(0.f, 0.f, 0.f, 0.f);
}

// ---------------------------------------------------------------------------
// Kernel 2: COO scatter-add into dense W^T [IN_F, OUT_F] (k-major), i.e.
// Wt[col][row] += val.  atomicAdd reproduces duplicate coalescing-by-sum.
// ---------------------------------------------------------------------------
__global__ void expander_scatter_kernel(const int* __restrict__ rows,
                                        const int* __restrict__ cols,
                                        const float* __restrict__ vals,
                                        float* __restrict__ Wt, int nnz) {
    int i = blockIdx.x * blockDim.x + threadIdx.x;
    if (i < nnz) {
        atomicAdd(&Wt[(size_t)cols[i] * OUT_F + rows[i]], vals[i]);
    }
}

// ---------------------------------------------------------------------------
// Kernel 3: C[N, OUT] = X[N, K] * Wt[K, OUT] with V_WMMA_F32_16X16X4_F32.
// Block: 256 threads (8 waves), tile 128x64, BK=32.
// Each wave: 2x2 WMMA tiles (32x32 region), waves arranged 4(m) x 2(n).
// LDS layout: row-major by m/n with k contiguous, stride 36 dwords
// (multiple of 4 for b128 stores; 36*m mod 64 spans 16 banks -> conflict-free
// b64 fragment reads).
// ---------------------------------------------------------------------------
#define SA 36
#define SB 36

__launch_bounds__(256, 2)
__global__ void expander_gemm_kernel(const float* __restrict__ X,
                                     const float* __restrict__ Wt,
                                     float* __restrict__ C) {
    __shared__ float As[128 * SA];
    __shared__ float Bs[64 * SB];

    const int tid  = threadIdx.x;
    const int wave = tid >> 5;
    const int lane = tid & 31;
    const int lm   = lane & 15;   // m/n within 16x16 tile
    const int lh   = lane >> 4;   // k-half selector (0 or 1)
    const int wm   = wave >> 1;   // 0..3 -> m offset wm*32
    const int wn   = wave & 1;    // 0..1 -> n offset wn*32

    const int m0 = blockIdx.y * 128;
    const int n0 = blockIdx.x * 64;

    // A staging: thread t loads rows r = (t>>3)+32p, cols (t&7)*4 .. +3
    const int arow = tid >> 3;         // 0..31
    const int acol = (tid & 7) << 2;   // 0,4,...,28
    // B staging: thread t loads n = t&63 at 8 consecutive k (kq group)
    const int bn  = tid & 63;
    const int bkq = tid >> 6;          // 0..3 -> k sub-block of 8

    v8f acc[2][2];
#pragma unroll
    for (int i = 0; i < 2; ++i)
#pragma unroll
        for (int j = 0; j < 2; ++j) acc[i][j] = v8f{};

    for (int k0 = 0; k0 < IN_F; k0 += 32) {
        // ---- global loads into registers (overlap with prior compute) ----
        float4 areg[4];
#pragma unroll
        for (int p = 0; p < 4; ++p) {
            const int r = arow + 32 * p;
            areg[p] = *(const float4*)(X + (size_t)(m0 + r) * IN_F + k0 + acol);
        }
        float breg[8];
#pragma unroll
        for (int i = 0; i < 8; ++i) {
            breg[i] = Wt[(size_t)(k0 + bkq * 8 + i) * OUT_F + n0 + bn];
        }

        __syncthreads();  // previous stage's LDS reads complete

        // ---- stage into LDS ----
#pragma unroll
        for (int p = 0; p < 4; ++p) {
            const int r = arow + 32 * p;
            *(float4*)(As + r * SA + acol) = areg[p];
        }
        *(float4*)(Bs + bn * SB + bkq * 8)     = make_float4(breg[0], breg[1], breg[2], breg[3]);
        *(float4*)(Bs + bn * SB + bkq * 8 + 4) = make_float4(breg[4], breg[5], breg[6], breg[7]);

        __syncthreads();

        // ---- 8 k-steps of 4, 2x2 WMMA tiles per wave ----
#pragma unroll
        for (int kk = 0; kk < 32; kk += 4) {
            const int kb = kk + 2 * lh;
            v2f af[2], bf[2];
#pragma unroll
            for (int mt = 0; mt < 2; ++mt)
                af[mt] = *(const v2f*)(As + (wm * 32 + mt * 16 + lm) * SA + kb);
#pragma unroll
            for (int nt = 0; nt < 2; ++nt)
                bf[nt] = *(const v2f*)(Bs + (wn * 32 + nt * 16 + lm) * SB + kb);

#pragma unroll
            for (int mt = 0; mt < 2; ++mt)
#pragma unroll
                for (int nt = 0; nt < 2; ++nt)
                    acc[mt][nt] = __builtin_amdgcn_wmma_f32_16x16x4_f32(
                        false, af[mt], false, bf[nt],
                        (short)0, acc[mt][nt], false, false);
        }
    }

    // ---- epilogue: C/D layout -> VGPR j holds M = j + 8*lh, N = lm ----
#pragma unroll
    for (int mt = 0; mt < 2; ++mt) {
#pragma unroll
        for (int nt = 0; nt < 2; ++nt) {
            const int col = n0 + wn * 32 + nt * 16 + lm;
#pragma unroll
            for (int j = 0; j < 8; ++j) {
                const int row = m0 + wm * 32 + mt * 16 + 8 * lh + j;
                C[(size_t)row * OUT_F + col] = acc[mt][nt][j];
            }
        }
    }
}

// ---------------------------------------------------------------------------
extern "C" void kernel_launch(void* const* d_in, const int* in_sizes, int n_in,
                              void* d_out, int out_size, void* d_ws, size_t ws_size,
                              hipStream_t stream) {
    const float* x    = (const float*)d_in[0];
    const int*   rows = (const int*)d_in[1];
    const int*   cols = (const int*)d_in[2];
    const float* vals = (const float*)d_in[3];
    float*       out  = (float*)d_out;
    float*       Wt   = (float*)d_ws;   // dense W^T [IN_F, OUT_F], 64 MB

    const int nnz = in_sizes[1];

    // 1) zero dense W^T (deterministic: re-done every call)
    expander_zero_kernel<<<4096, 256, 0, stream>>>((float4*)Wt,
                                                   (IN_F * OUT_F) / 4);
    // 2) COO scatter-add
    expander_scatter_kernel<<<(nnz + 255) / 256, 256, 0, stream>>>(
        rows, cols, vals, Wt, nnz);
    // 3) WMMA f32 GEMM: grid.x = n-tiles (fast) so blocks sharing an X stripe
    //    run adjacently; W^T stays resident in the 192MB L2.
    dim3 grid(OUT_F / 64, NROWS / 128);
    expander_gemm_kernel<<<grid, 256, 0, stream>>>(x, Wt, out);
}